// DTransformer_40321152975253
// MI455X (gfx1250) — compile-verified
//
#include <hip/hip_runtime.h>
#include <hip/hip_bf16.h>

#define Bb 4
#define Ss 256
#define Dd 256
#define Hh 8
#define Kk 16
#define DH 32
#define STRIP 64
#define SCPAD 257   // 256 + 1 pad -> phase-3 A-frag column reads are bank-conflict-free

typedef float v2f __attribute__((ext_vector_type(2)));
typedef float v8f __attribute__((ext_vector_type(8)));

__device__ __forceinline__ v8f wmma_f32x4(v2f a, v2f b, v8f c) {
    // D = A(16x4 f32) * B(4x16 f32) + C(16x16 f32)
    return __builtin_amdgcn_wmma_f32_16x16x4_f32(false, a, false, b, (short)0, c, false, false);
}

__device__ __forceinline__ float wave_sum(float v) {
    #pragma unroll
    for (int o = 16; o; o >>= 1) v += __shfl_xor(v, o, 32);
    return v;
}
__device__ __forceinline__ float wave_max(float v) {
    #pragma unroll
    for (int o = 16; o; o >>= 1) v = fmaxf(v, __shfl_xor(v, o, 32));
    return v;
}

// ---------------------------------------------------------------------------
// Y[row,col] = sum_k X[row,k] * W[col,k] + bias[col] (+ residual)
// res_mode: 0 = none, 1 = R[row,col], 2 = kparams[(row/S)%K, col]
// One wave computes one 16x16 tile; K-loop of D/4 = 64 WMMAs.
// ---------------------------------------------------------------------------
__global__ __launch_bounds__(256) void gemm_xwT(
    const float* __restrict__ X, const float* __restrict__ W,
    const float* __restrict__ bias, const float* __restrict__ R, int res_mode,
    const float* __restrict__ kparams, float* __restrict__ Y, int rows)
{
    const int wave = threadIdx.x >> 5, lane = threadIdx.x & 31;
    const int lr = lane & 15, lh = lane >> 4;
    const int tile = blockIdx.x * 8 + wave;
    const int nt_per = Dd / 16;
    const int mt = tile / nt_per, nt = tile % nt_per;
    const int M0 = mt * 16, N0 = nt * 16;
    if (M0 >= rows) return;

    const float* xrow = X + (size_t)(M0 + lr) * Dd;
    const float* wrow = W + (size_t)(N0 + lr) * Dd;
    v8f acc = {};
    for (int k0 = 0; k0 < Dd; k0 += 4) {
        v2f a, b;
        a.x = xrow[k0 + lh * 2 + 0];
        a.y = xrow[k0 + lh * 2 + 1];
        b.x = wrow[k0 + lh * 2 + 0];   // B[k][n] = W[n][k]  (Y = X * W^T)
        b.y = wrow[k0 + lh * 2 + 1];
        acc = wmma_f32x4(a, b, acc);
    }
    const int col = N0 + lr;
    const float bv = bias ? bias[col] : 0.0f;
    #pragma unroll
    for (int v = 0; v < 8; ++v) {
        const int row = M0 + v + lh * 8;
        float r = acc[v] + bv;
        if (res_mode == 1)      r += R[(size_t)row * Dd + col];
        else if (res_mode == 2) r += kparams[(size_t)((row / Ss) % Kk) * Dd + col];
        Y[(size_t)row * Dd + col] = r;
    }
}

// ---------------------------------------------------------------------------
// Fused attention for one (head h, sequence n): processes 64-row strips.
//   phase1: scores = Q K^T / sqrt(DH)          (WMMA -> LDS)
//   phase2: softmax -> distance decay -> softmax -> maxout, wave-per-row:
//           8 cols/lane in registers, shfl reductions + shfl_up prefix scan,
//           coalesced 1KB-per-wave global score rows ([B,H,S,K,S] transpose
//           folded into the store index for k_scores)
//   phase3: O = scores @ V                     (WMMA from LDS)
// q_bcast: Q row is kparams projection PK[n%K] (constant over seq position)
// kv_div : K/V batch index = n / kv_div (layer-4 broadcast over K)
// strict : causal mask j<q (else j<=q). score_mode: 0 none, 1 [n,h,q,j], 2 k_scores
// ---------------------------------------------------------------------------
__global__ __launch_bounds__(256) void attn_kernel(
    const float* __restrict__ Qp, const float* __restrict__ Kp,
    const float* __restrict__ Vp, const float* __restrict__ gam,
    float* __restrict__ Oout, float* __restrict__ sc_out,
    int q_bcast, int kv_div, int strict, int maxout, int score_mode)
{
    __shared__ float sc[STRIP][SCPAD];
    const int h = blockIdx.x, n = blockIdx.y;
    const int wave = threadIdx.x >> 5, lane = threadIdx.x & 31;
    const int lr = lane & 15, lh = lane >> 4;
    const int nkv = n / kv_div;
    const float* Kbase = Kp + ((size_t)nkv * Ss) * Dd + h * DH;
    const float* Vbase = Vp + ((size_t)nkv * Ss) * Dd + h * DH;
    const float ghead = -fabsf(gam[h]);
    const float scale = 0.17677669529663687f;   // 1/sqrt(32)

    for (int r0 = 0; r0 < Ss; r0 += STRIP) {
        // ---- phase 1: scores strip [STRIP x S] ----
        {
            const int rt = wave >> 1, ch = wave & 1;
            const int Mrow = r0 + rt * 16 + lr;
            const float* qrow = q_bcast
                ? (Qp + (size_t)(n % Kk) * Dd + h * DH)
                : (Qp + ((size_t)n * Ss + Mrow) * Dd + h * DH);
            for (int ntile = 0; ntile < 8; ++ntile) {
                const int N0 = ch * 128 + ntile * 16;
                const float* krow = Kbase + (size_t)(N0 + lr) * Dd;
                v8f acc = {};
                #pragma unroll
                for (int k0 = 0; k0 < DH; k0 += 4) {
                    v2f a, b;
                    a.x = qrow[k0 + lh * 2 + 0];
                    a.y = qrow[k0 + lh * 2 + 1];
                    b.x = krow[k0 + lh * 2 + 0];   // B[k][j] = K[j][k]
                    b.y = krow[k0 + lh * 2 + 1];
                    acc = wmma_f32x4(a, b, acc);
                }
                #pragma unroll
                for (int v = 0; v < 8; ++v)
                    sc[rt * 16 + v + lh * 8][N0 + lr] = acc[v] * scale;
            }
        }
        __syncthreads();

        // ---- phase 2: wave-per-row softmax / decay / softmax / maxout ----
        for (int rr = wave; rr < STRIP; rr += 8) {
            const int q = r0 + rr;
            float* row = sc[rr];
            const int lim = strict ? q : (q + 1);      // j < lim is unmasked
            const int jbase = lane * 8;                // 8 contiguous cols per lane
            float x[8];
            #pragma unroll
            for (int i = 0; i < 8; ++i) {
                const int j = jbase + i;
                x[i] = (j < lim) ? row[j] : -1e32f;
            }
            // softmax #1 statistics
            float m1l = -3.4e38f;
            #pragma unroll
            for (int i = 0; i < 8; ++i) m1l = fmaxf(m1l, x[i]);
            const float m1 = wave_max(m1l);
            float e[8];
            float lsum = 0.0f;
            #pragma unroll
            for (int i = 0; i < 8; ++i) { e[i] = __expf(x[i] - m1); lsum += e[i]; }
            const float inv1 = 1.0f / wave_sum(lsum);
            // exclusive prefix of per-lane sums (wave inclusive scan via shfl_up)
            float pre = lsum;
            #pragma unroll
            for (int off = 1; off < 32; off <<= 1) {
                const float t = __shfl_up(pre, off, 32);
                if (lane >= off) pre += t;
            }
            pre -= lsum;
            // cumsum -> distance decay -> reweighted logits
            float run = pre;
            float y[8];
            float m2l = -3.4e38f;
            #pragma unroll
            for (int i = 0; i < 8; ++i) {
                run += e[i];
                const int j = jbase + i;
                const float cum = run * inv1;          // cumsum of softmax #1
                const float pos = fabsf((float)(q - j));
                const float dist = sqrtf(fmaxf((1.0f - cum) * pos, 0.0f));
                float eff = __expf(dist * ghead);
                eff = fminf(fmaxf(eff, 1e-5f), 1e5f);
                y[i] = (j < lim) ? x[i] * eff : -1e32f;
                m2l = fmaxf(m2l, y[i]);
            }
            // softmax #2
            const float m2 = wave_max(m2l);
            float l2 = 0.0f;
            #pragma unroll
            for (int i = 0; i < 8; ++i) { y[i] = __expf(y[i] - m2); l2 += y[i]; }
            const float inv2 = 1.0f / wave_sum(l2);
            float mxl = 0.0f;
            #pragma unroll
            for (int i = 0; i < 8; ++i) {
                const int j = jbase + i;
                y[i] = (j < lim) ? y[i] * inv2 : 0.0f;
                mxl = fmaxf(mxl, y[i]);
            }
            const float smul = maxout ? fminf(1.0f / fmaxf(wave_max(mxl), 1e-8f), 5.0f)
                                      : 1.0f;
            size_t base = 0;
            if (score_mode == 1) {
                base = (((size_t)n * Hh + h) * Ss + q) * Ss;
            } else if (score_mode == 2) {
                const int b = n / Kk, ki = n % Kk;      // [B,H,S,K,S]
                base = ((((size_t)b * Hh + h) * Ss + q) * Kk + ki) * Ss;
            }
            #pragma unroll
            for (int i = 0; i < 8; ++i) {
                const float p = y[i] * smul;
                row[jbase + i] = p;
                if (score_mode) sc_out[base + jbase + i] = p;   // coalesced per wave
            }
        }
        __syncthreads();

        // ---- phase 3: O strip = scores @ V ----
        {
            const int rt = wave >> 1, nh = wave & 1;
            const int N0 = nh * 16;
            v8f acc = {};
            for (int k0 = 0; k0 < Ss; k0 += 4) {
                const int kk = k0 + lh * 2;
                v2f a, b;
                a.x = sc[rt * 16 + lr][kk + 0];
                a.y = sc[rt * 16 + lr][kk + 1];
                const float* v0 = Vbase + (size_t)kk * Dd + N0 + lr;
                b.x = v0[0];                 // V[kk]  [N0+lr]
                b.y = v0[Dd];                // V[kk+1][N0+lr]
                acc = wmma_f32x4(a, b, acc);
            }
            #pragma unroll
            for (int v = 0; v < 8; ++v) {
                const int row = r0 + rt * 16 + v + lh * 8;
                Oout[((size_t)n * Ss + row) * Dd + h * DH + N0 + lr] = acc[v];
            }
        }
        __syncthreads();
    }
}

// ---------------------------------------------------------------------------
// LayerNorm over D=256, one wave (32 lanes x 8 elems) per row, shfl reductions.
// zmode=1 writes with the [B,K,S,D] -> [B,S,K*D] transpose folded in.
// ---------------------------------------------------------------------------
__global__ __launch_bounds__(256) void ln_kernel(
    const float* __restrict__ Xin, const float* __restrict__ w,
    const float* __restrict__ bb, float* __restrict__ Yout, int rows, int zmode)
{
    const int wave = threadIdx.x >> 5, lane = threadIdx.x & 31;
    const int row = blockIdx.x * 8 + wave;
    if (row >= rows) return;
    const float* x = Xin + (size_t)row * Dd;
    float v[8];
    float s = 0.0f;
    #pragma unroll
    for (int i = 0; i < 8; ++i) { v[i] = x[lane + i * 32]; s += v[i]; }
    s = wave_sum(s);
    const float mu = s * (1.0f / Dd);
    float s2 = 0.0f;
    #pragma unroll
    for (int i = 0; i < 8; ++i) { const float d = v[i] - mu; s2 += d * d; }
    s2 = wave_sum(s2);
    const float inv = rsqrtf(s2 * (1.0f / Dd) + 1e-5f);
    size_t obase;
    if (zmode) {
        const int nidx = row / Ss, sidx = row % Ss;
        const int b = nidx / Kk, ki = nidx % Kk;
        obase = (((size_t)b * Ss + sidx) * Kk + ki) * Dd;
    } else {
        obase = (size_t)row * Dd;
    }
    #pragma unroll
    for (int i = 0; i < 8; ++i) {
        const int c = lane + i * 32;
        Yout[obase + c] = (v[i] - mu) * inv * w[c] + bb[c];
    }
}

// ---------------------------------------------------------------------------
extern "C" void kernel_launch(void* const* d_in, const int* in_sizes, int n_in,
                              void* d_out, int out_size, void* d_ws, size_t ws_size,
                              hipStream_t stream)
{
    (void)in_sizes; (void)n_in; (void)out_size; (void)ws_size;
    const float* emb_con = (const float*)d_in[0];
    const float* emb_int = (const float*)d_in[1];
    // d_in[2] = seqs_length (all S, unused)
    const float* Wq1 = (const float*)d_in[3],  *bq1 = (const float*)d_in[4];
    const float* Wv1 = (const float*)d_in[5],  *bv1 = (const float*)d_in[6];
    const float* Wo1 = (const float*)d_in[7],  *bo1 = (const float*)d_in[8];
    const float* g1  = (const float*)d_in[9];
    const float* lnw1= (const float*)d_in[10], *lnb1= (const float*)d_in[11];
    const float* Wq2 = (const float*)d_in[12], *bq2 = (const float*)d_in[13];
    const float* Wv2 = (const float*)d_in[14], *bv2 = (const float*)d_in[15];
    const float* Wo2 = (const float*)d_in[16], *bo2 = (const float*)d_in[17];
    const float* g2  = (const float*)d_in[18];
    const float* lnw2= (const float*)d_in[19], *lnb2= (const float*)d_in[20];
    const float* Wq4 = (const float*)d_in[21], *bq4 = (const float*)d_in[22];
    const float* Wk4 = (const float*)d_in[23], *bk4 = (const float*)d_in[24];
    const float* Wv4 = (const float*)d_in[25], *bv4 = (const float*)d_in[26];
    const float* Wo4 = (const float*)d_in[27], *bo4 = (const float*)d_in[28];
    const float* g4  = (const float*)d_in[29];
    const float* lnw4= (const float*)d_in[30], *lnb4= (const float*)d_in[31];
    const float* kpar= (const float*)d_in[32];

    float* ws = (float*)d_ws;
    const size_t sz  = (size_t)Bb * Ss * Dd;       // 262144
    const size_t big = (size_t)Bb * Kk * Ss * Dd;  // 4194304
    float* Qp  = ws + 0 * sz;
    float* Vp  = ws + 1 * sz;
    float* Ob  = ws + 2 * sz;
    float* Xb  = ws + 3 * sz;
    float* IR  = ws + 4 * sz;
    float* KR  = ws + 5 * sz;
    float* K4  = ws + 6 * sz;
    float* V4  = ws + 7 * sz;
    float* PK4 = ws + 8 * sz;
    float* O4  = ws + 9 * sz;
    float* X4  = O4 + big;

    float* z_out  = (float*)d_out;
    float* qsc    = z_out + big;                               // [B,H,S,S]
    float* ksc    = qsc + (size_t)Bb * Hh * Ss * Ss;           // [B,H,S,K,S]

    const int rowsS  = Bb * Ss;         // 1024
    const int rowsBK = Bb * Kk * Ss;    // 16384
    const int gS  = (rowsS  / 16) * (Dd / 16) / 8;   // 128 blocks
    const int gBK = (rowsBK / 16) * (Dd / 16) / 8;   // 2048 blocks
    const dim3 blk(256);

    // ---- layer 1 (q=k=v=emb_interaction, key linear == query linear) ----
    gemm_xwT<<<gS, blk, 0, stream>>>(emb_int, Wq1, bq1, nullptr, 0, nullptr, Qp, rowsS);
    gemm_xwT<<<gS, blk, 0, stream>>>(emb_int, Wv1, bv1, nullptr, 0, nullptr, Vp, rowsS);
    attn_kernel<<<dim3(Hh, Bb), blk, 0, stream>>>(Qp, Qp, Vp, g1, Ob, nullptr,
                                                  0, 1, /*strict=*/0, /*maxout=*/0, /*score=*/0);
    gemm_xwT<<<gS, blk, 0, stream>>>(Ob, Wo1, bo1, emb_int, 1, nullptr, Xb, rowsS);
    ln_kernel<<<rowsS / 8, blk, 0, stream>>>(Xb, lnw1, lnb1, IR, rowsS, 0);

    // ---- layer 2 (q=k=emb_concept, v=interaction_repr) ----
    gemm_xwT<<<gS, blk, 0, stream>>>(emb_con, Wq2, bq2, nullptr, 0, nullptr, Qp, rowsS);
    gemm_xwT<<<gS, blk, 0, stream>>>(IR,      Wv2, bv2, nullptr, 0, nullptr, Vp, rowsS);
    attn_kernel<<<dim3(Hh, Bb), blk, 0, stream>>>(Qp, Qp, Vp, g2, Ob, qsc,
                                                  0, 1, /*strict=*/0, /*maxout=*/0, /*score=*/1);
    gemm_xwT<<<gS, blk, 0, stream>>>(Ob, Wo2, bo2, emb_con, 1, nullptr, Xb, rowsS);
    ln_kernel<<<rowsS / 8, blk, 0, stream>>>(Xb, lnw2, lnb2, KR, rowsS, 0);

    // ---- layer 4 (Q = knowledge_params broadcast; K/V broadcast over K) ----
    gemm_xwT<<<2,  blk, 0, stream>>>(kpar,    Wq4, bq4, nullptr, 0, nullptr, PK4, Kk);
    gemm_xwT<<<gS, blk, 0, stream>>>(emb_con, Wk4, bk4, nullptr, 0, nullptr, K4,  rowsS);
    gemm_xwT<<<gS, blk, 0, stream>>>(KR,      Wv4, bv4, nullptr, 0, nullptr, V4,  rowsS);
    attn_kernel<<<dim3(Hh, Bb * Kk), blk, 0, stream>>>(PK4, K4, V4, g4, O4, ksc,
                                                  /*q_bcast=*/1, /*kv_div=*/Kk,
                                                  /*strict=*/1, /*maxout=*/1, /*score=*/2);
    gemm_xwT<<<gBK, blk, 0, stream>>>(O4, Wo4, bo4, nullptr, 2, kpar, X4, rowsBK);
    ln_kernel<<<rowsBK / 8, blk, 0, stream>>>(X4, lnw4, lnb4, z_out, rowsBK, /*zmode=*/1);
}